// EquivariantModule_76897094467617
// MI455X (gfx1250) — compile-verified
//
#include <hip/hip_runtime.h>

typedef __attribute__((ext_vector_type(2))) float v2f;
typedef __attribute__((ext_vector_type(8))) float v8f;

#define R_CUT2 0.25f
#define S3  1.7320508075688772f
#define S5  2.23606797749979f
#define S15 3.872983346207417f

// One wave (32 lanes) computes one 16x16 tile of the pair matrix.
// d2 tile comes from a single v_wmma_f32_16x16x4_f32:
//   A row i   = ( pix,  piy,  piz, |pi|^2 )   (16x4, f32)
//   B col j   = (-2pjx,-2pjy,-2pjz, 1     )   (4x16, f32)
//   C[:, j]   = |pj|^2 (broadcast per column)
//   D[i, j]   = |pi|^2 + |pj|^2 - 2 pi.pj = dist^2
__global__ __launch_bounds__(256) void pair_kernel(
    const float* __restrict__ pos,   // [B,N,3]
    float*       __restrict__ r_out, // [B,N,N]
    float*       __restrict__ sh_out,// [B,N,N,9]
    int N, int tilesPerB, int jTiles)
{
    const int wave = threadIdx.x >> 5;
    const int lane = threadIdx.x & 31;
    const int tile = blockIdx.x * 8 + wave;
    const int b    = tile / tilesPerB;
    const int rem  = tile - b * tilesPerB;
    const int i0   = (rem / jTiles) << 4;
    const int j0   = (rem % jTiles) << 4;

    const float* pb = pos + (size_t)b * N * 3;

    const int half = lane >> 4;   // 0: K=0,1 half  1: K=2,3 half
    const int sl   = lane & 15;   // row/col index within tile

    // ---- A operand: position rows i0+sl (ISA 32-bit A 16x4 layout) ----
    const float* pa = pb + (size_t)(i0 + sl) * 3;
    const float pax = pa[0], pay = pa[1], paz = pa[2];
    const float na  = fmaf(pax, pax, fmaf(pay, pay, paz * paz));
    v2f A;
    A.x = half ? paz : pax;       // VGPR0: K=0 (lo half) / K=2 (hi half)
    A.y = half ? na  : pay;       // VGPR1: K=1 (lo half) / K=3 (hi half)

    // ---- B operand: position cols j0+sl (mirrored layout, N = lane%16) ----
    const float* pq = pb + (size_t)(j0 + sl) * 3;
    const float pjx = pq[0], pjy = pq[1], pjz = pq[2];
    const float nq  = fmaf(pjx, pjx, fmaf(pjy, pjy, pjz * pjz));
    v2f Bv;
    Bv.x = half ? (-2.0f * pjz) : (-2.0f * pjx);
    Bv.y = half ? 1.0f          : (-2.0f * pjy);

    // ---- C preloaded with |pj|^2 (constant down each column) ----
    v8f C = {nq, nq, nq, nq, nq, nq, nq, nq};

    // D = A*B + C  ->  full 16x16 tile of squared distances
    C = __builtin_amdgcn_wmma_f32_16x16x4_f32(
            /*neg_a=*/false, A, /*neg_b=*/false, Bv,
            /*c_mod=*/(short)0, C, /*reuse_a=*/false, /*reuse_b=*/false);

    const int j = j0 + sl;
    #pragma unroll
    for (int m = 0; m < 8; ++m) {
        const int M = m + (half << 3);   // C/D layout: hi half holds M=8..15
        const int i = i0 + M;

        // edge vector src->dst (pos[j] - pos[i]); tiny broadcast loads, L1-resident
        const float* pi = pb + (size_t)i * 3;
        const float relx = pjx - pi[0];
        const float rely = pjy - pi[1];
        const float relz = pjz - pi[2];

        float d2 = fmaxf(C[m], 0.0f);                    // clamp rounding residue
        const bool  on  = (d2 < R_CUT2) && (i != j);     // dist<0.5, no self-loop
        const float msk = on ? 1.0f : 0.0f;
        const float s   = (d2 > 0.0f) ? __builtin_amdgcn_rsqf(d2) : 0.0f;
        const float dist = d2 * s;                       // d2 * rsq(d2) = sqrt(d2)
        const float ux = relx * s, uy = rely * s, uz = relz * s;
        const float x2 = ux * ux, y2 = uy * uy, z2 = uz * uz;

        const size_t e = ((size_t)b * N + (size_t)i) * N + (size_t)j;

        // streamed once, never re-read -> non-temporal stores
        __builtin_nontemporal_store(on ? dist : 0.0f, r_out + e);

        float* hp = sh_out + e * 9;
        __builtin_nontemporal_store(msk,                          hp + 0);
        __builtin_nontemporal_store(S3 * ux * msk,                hp + 1);
        __builtin_nontemporal_store(S3 * uy * msk,                hp + 2);
        __builtin_nontemporal_store(S3 * uz * msk,                hp + 3);
        __builtin_nontemporal_store(S15 * ux * uz * msk,          hp + 4);
        __builtin_nontemporal_store(S15 * ux * uy * msk,          hp + 5);
        __builtin_nontemporal_store(S5 * (y2 - 0.5f * (x2 + z2)) * msk, hp + 6);
        __builtin_nontemporal_store(S15 * uy * uz * msk,          hp + 7);
        __builtin_nontemporal_store(0.5f * S15 * (z2 - x2) * msk, hp + 8);
    }
}

// readout[b,n] = x[b,n,:] . W + bias   (trivial: 8192 outputs)
__global__ __launch_bounds__(256) void readout_kernel(
    const float* __restrict__ x, const float* __restrict__ W,
    const float* __restrict__ bias, float* __restrict__ out, int total)
{
    const int t = blockIdx.x * blockDim.x + threadIdx.x;
    if (t >= total) return;
    const float* xr = x + (size_t)t * 12;
    float acc = bias[0];
    #pragma unroll
    for (int f = 0; f < 12; ++f) acc = fmaf(xr[f], W[f], acc);
    out[t] = acc;
}

extern "C" void kernel_launch(void* const* d_in, const int* in_sizes, int n_in,
                              void* d_out, int out_size, void* d_ws, size_t ws_size,
                              hipStream_t stream)
{
    const float* pos  = (const float*)d_in[0];  // [4,2048,3]
    const float* x    = (const float*)d_in[1];  // [4,2048,12]
    const float* W    = (const float*)d_in[2];  // [1,12]
    const float* bias = (const float*)d_in[3];  // [1]

    const int B = 4, N = 2048;
    float* out     = (float*)d_out;
    float* readout = out;                                   // B*N
    float* r_out   = out + (size_t)B * N;                   // B*N*N
    float* sh_out  = r_out + (size_t)B * N * N;             // B*N*N*9

    readout_kernel<<<(B * N + 255) / 256, 256, 0, stream>>>(x, W, bias, readout, B * N);

    const int jTiles     = N / 16;            // 128
    const int tilesPerB  = jTiles * jTiles;   // 16384
    const int totalTiles = B * tilesPerB;     // 65536 tiles, 8 waves/block
    pair_kernel<<<totalTiles / 8, 256, 0, stream>>>(pos, r_out, sh_out, N, tilesPerB, jTiles);
}